// APA_3607772528654
// MI455X (gfx1250) — compile-verified
//
#include <hip/hip_runtime.h>

typedef __attribute__((ext_vector_type(2))) float v2f;
typedef __attribute__((ext_vector_type(8))) float v8f;

#define DF 64
#define NUM_ITER 10

static __device__ __forceinline__ int imin(int a, int b) { return a < b ? a : b; }

// ---------------------------------------------------------------------------
// Degree histogram: deg[n] = #edges with row==n
__global__ void count_rows(const int* __restrict__ row, int* __restrict__ cnt, int E) {
    int e = blockIdx.x * blockDim.x + threadIdx.x;
    if (e < E) atomicAdd(&cnt[row[e]], 1);
}

// dis[n] = deg>0 ? rsqrt(deg) : 0   (deg>=1 integer => max(deg,1)==deg)
__global__ void calc_dis(const int* __restrict__ cnt, float* __restrict__ dis, int N) {
    int n = blockIdx.x * blockDim.x + threadIdx.x;
    if (n < N) {
        int d = cnt[n];
        dis[n] = (d > 0) ? rsqrtf((float)d) : 0.0f;
    }
}

// ---------------------------------------------------------------------------
// Exclusive scan of cnt -> rowptr  (3 phases; N=100000, 391 blocks of 256)
__global__ void scan_phase1(const int* __restrict__ cnt, int* __restrict__ excl,
                            int* __restrict__ bsum, int N) {
    __shared__ int s[256];
    int t = threadIdx.x;
    int i = blockIdx.x * 256 + t;
    int v = (i < N) ? cnt[i] : 0;
    s[t] = v;
    __syncthreads();
    for (int o = 1; o < 256; o <<= 1) {
        int x = (t >= o) ? s[t - o] : 0;
        __syncthreads();
        s[t] += x;
        __syncthreads();
    }
    if (i < N) excl[i] = s[t] - v;
    if (t == 255) bsum[blockIdx.x] = s[t];
}

__global__ void scan_phase2(int* __restrict__ bsum, int nb) {
    __shared__ int s[512];
    int t = threadIdx.x;
    int v = (t < nb) ? bsum[t] : 0;
    s[t] = v;
    __syncthreads();
    for (int o = 1; o < 512; o <<= 1) {
        int x = (t >= o) ? s[t - o] : 0;
        __syncthreads();
        s[t] += x;
        __syncthreads();
    }
    if (t < nb) bsum[t] = s[t] - v;   // exclusive block offsets
}

__global__ void scan_phase3(int* __restrict__ excl, const int* __restrict__ bsum,
                            int N, int E) {
    int i = blockIdx.x * blockDim.x + threadIdx.x;
    if (i < N) excl[i] += bsum[i >> 8];
    if (i == 0) excl[N] = E;
}

__global__ void copy_pos(const int* __restrict__ rowptr, int* __restrict__ pos, int N) {
    int i = blockIdx.x * blockDim.x + threadIdx.x;
    if (i < N) pos[i] = rowptr[i];
}

// Counting-sort scatter: place (col, w) of each edge into its row's CSR slot.
__global__ void scatter_edges(const int* __restrict__ row, const int* __restrict__ col,
                              const float* __restrict__ dis, int* __restrict__ pos,
                              int* __restrict__ cols_s, float* __restrict__ ws_s, int E) {
    int e = blockIdx.x * blockDim.x + threadIdx.x;
    if (e >= E) return;
    int r = row[e], c = col[e];
    float w = (r == c) ? 0.0f : dis[r] * dis[c];
    int p = atomicAdd(&pos[r], 1);
    cols_s[p] = c;
    ws_s[p]   = w;
}

// Masked reset: out[mask[i]] = x[mask[i]]  (duplicate ids are idempotent)
__global__ void set_known(const float* __restrict__ x, const int* __restrict__ mask,
                          float* __restrict__ out, int K) {
    int i = blockIdx.x * blockDim.x + threadIdx.x;
    if (i >= K * DF) return;
    int node = mask[i >> 6];
    int d = i & 63;
    out[node * DF + d] = x[node * DF + d];
}

// ---------------------------------------------------------------------------
// CSR SpMM via V_WMMA_F32_16X16X4_F32.
// One wave per 16-row tile; 4 f32 16x16 accumulators cover the 64 features.
// Per chunk of 4 CSR-consecutive edges:
//   A (16x4, MxK): A[m][k] = w[e_k] iff edge e_k belongs to row r0+m (one-hot),
//     lane layout per ISA: lanes 0-15 hold (K0,K1), lanes 16-31 hold (K2,K3).
//   B (4x16, KxN): B[k][n] = src[col[e_k]][nbase+n].
//     K layout assumed to mirror the documented C layout:
//     VGPR0 -> K0 (lanes 0-15) / K2 (lanes 16-31); VGPR1 -> K1 / K3.
// Tail edges padded with w=0 (A zero) and clamped col loads.
// All per-lane guards are branch-free (unsigned range compare -> v_cndmask),
// so EXEC stays all-ones through the WMMA loop.
__global__ __launch_bounds__(256) void spmm_wmma(
    const float* __restrict__ src, float* __restrict__ dst,
    const int* __restrict__ rowptr, const int* __restrict__ cols,
    const float* __restrict__ wvals, int N) {
    int gtid = blockIdx.x * blockDim.x + threadIdx.x;
    int wave = gtid >> 5;
    int lane = threadIdx.x & 31;
    int ntiles = (N + 15) >> 4;
    if (wave >= ntiles) return;           // wave-uniform: EXEC stays all-ones

    int r0 = wave << 4;
    int m  = lane & 15;                   // M index (A) / N index (B,C)
    int hi = lane >> 4;                   // half-wave select
    int rIdx = imin(r0 + m, N - 1);
    int rpm  = rowptr[rIdx];
    unsigned span = (unsigned)(rowptr[rIdx + 1] - rpm);   // edges of row r0+m
    int beg  = rowptr[r0];
    int end  = rowptr[imin(r0 + 16, N)];
    int endm1 = end - 1;

    v8f c0 = {}, c1 = {}, c2 = {}, c3 = {};
    int k0 = hi * 2;                      // this lane's K slots: k0, k0+1

#pragma unroll 2
    for (int base = beg; base < end; base += 4) {
        int e0 = base + k0;
        int e1 = e0 + 1;
        int ec0 = imin(e0, endm1);        // clamp for tail (A is 0 there)
        int ec1 = imin(e1, endm1);
        float w0 = wvals[ec0];
        float w1 = wvals[ec1];
        int o0 = cols[ec0] * DF + m;      // 32-bit element offsets (N*DF < 2^31)
        int o1 = cols[ec1] * DF + m;
        // branch-free one-hot: edge belongs to this lane's row?
        bool in0 = (unsigned)(e0 - rpm) < span;
        bool in1 = (unsigned)(e1 - rpm) < span;
        v2f a;
        a.x = in0 ? w0 : 0.0f;
        a.y = in1 ? w1 : 0.0f;
        v2f b;
        b.x = src[o0];      b.y = src[o1];
        c0 = __builtin_amdgcn_wmma_f32_16x16x4_f32(false, a, false, b, (short)0, c0, false, false);
        b.x = src[o0 + 16]; b.y = src[o1 + 16];
        c1 = __builtin_amdgcn_wmma_f32_16x16x4_f32(false, a, false, b, (short)0, c1, false, false);
        b.x = src[o0 + 32]; b.y = src[o1 + 32];
        c2 = __builtin_amdgcn_wmma_f32_16x16x4_f32(false, a, false, b, (short)0, c2, false, false);
        b.x = src[o0 + 48]; b.y = src[o1 + 48];
        c3 = __builtin_amdgcn_wmma_f32_16x16x4_f32(false, a, false, b, (short)0, c3, false, false);
    }

    // C layout: VGPR v -> M = v (lanes 0-15) / v+8 (lanes 16-31), N = lane%16
    int rowBase = r0 + hi * 8;
    if (r0 + 16 <= N) {                   // wave-uniform fast path (always for N%16==0)
#pragma unroll
        for (int v = 0; v < 8; ++v) {
            float* drow = dst + (rowBase + v) * DF + m;
            drow[0]  = c0[v];
            drow[16] = c1[v];
            drow[32] = c2[v];
            drow[48] = c3[v];
        }
    } else {
#pragma unroll
        for (int v = 0; v < 8; ++v) {
            int rr = rowBase + v;
            if (rr < N) {
                float* drow = dst + rr * DF + m;
                drow[0]  = c0[v];
                drow[16] = c1[v];
                drow[32] = c2[v];
                drow[48] = c3[v];
            }
        }
    }
}

// ---------------------------------------------------------------------------
extern "C" void kernel_launch(void* const* d_in, const int* in_sizes, int n_in,
                              void* d_out, int out_size, void* d_ws, size_t ws_size,
                              hipStream_t stream) {
    const float* x    = (const float*)d_in[0];
    const int*   ei   = (const int*)d_in[1];
    const int*   mask = (const int*)d_in[2];
    int N = in_sizes[0] / DF;    // 100000
    int E = in_sizes[1] / 2;     // 1600000
    int K = in_sizes[2];         // 40000
    const int* row = ei;
    const int* col = ei + E;

    // carve workspace (~40 MB)
    char* p = (char*)d_ws;
    auto carve = [&](size_t bytes) -> void* {
        void* r = (void*)p;
        p += (bytes + 255) & ~(size_t)255;
        return r;
    };
    int*   cnt    = (int*)  carve(sizeof(int) * (size_t)N);
    int*   rowptr = (int*)  carve(sizeof(int) * (size_t)(N + 1));
    int*   pos    = (int*)  carve(sizeof(int) * (size_t)N);
    int*   bsum   = (int*)  carve(sizeof(int) * 1024);
    float* dis    = (float*)carve(sizeof(float) * (size_t)N);
    int*   cols_s = (int*)  carve(sizeof(int) * (size_t)E);
    float* ws_s   = (float*)carve(sizeof(float) * (size_t)E);
    float* buf    = (float*)carve(sizeof(float) * (size_t)N * DF);
    float* out    = (float*)d_out;

    const int tB = 256;
    hipMemsetAsync(cnt, 0, sizeof(int) * (size_t)N, stream);
    count_rows<<<(E + tB - 1) / tB, tB, 0, stream>>>(row, cnt, E);
    calc_dis  <<<(N + tB - 1) / tB, tB, 0, stream>>>(cnt, dis, N);

    int nb = (N + 255) / 256;    // 391 <= 512
    scan_phase1<<<nb, 256, 0, stream>>>(cnt, rowptr, bsum, N);
    scan_phase2<<<1, 512, 0, stream>>>(bsum, nb);
    scan_phase3<<<(N + tB - 1) / tB, tB, 0, stream>>>(rowptr, bsum, N, E);
    copy_pos   <<<(N + tB - 1) / tB, tB, 0, stream>>>(rowptr, pos, N);
    scatter_edges<<<(E + tB - 1) / tB, tB, 0, stream>>>(row, col, dis, pos, cols_s, ws_s, E);

    // out0 = zeros with known rows set
    hipMemsetAsync(out, 0, sizeof(float) * (size_t)N * DF, stream);
    set_known<<<(K * DF + tB - 1) / tB, tB, 0, stream>>>(x, mask, out, K);

    int ntiles = (N + 15) / 16;
    int waves  = ntiles * 32;
    int blocks = (waves + 255) / 256;

    float* cur = out;
    float* nxt = buf;
    for (int it = 0; it < NUM_ITER; ++it) {
        spmm_wmma<<<blocks, 256, 0, stream>>>(cur, nxt, rowptr, cols_s, ws_s, N);
        set_known<<<(K * DF + tB - 1) / tB, tB, 0, stream>>>(x, mask, nxt, K);
        float* t = cur; cur = nxt; nxt = t;
    }
    // NUM_ITER even -> cur == d_out; keep a safety copy for odd iteration counts
    if (cur != out) {
        hipMemcpyAsync(out, cur, sizeof(float) * (size_t)N * DF,
                       hipMemcpyDeviceToDevice, stream);
    }
    (void)n_in; (void)ws_size; (void)out_size;
}